// RasterizePointsXYsBlending_82506321756355
// MI455X (gfx1250) — compile-verified
//
#include <hip/hip_runtime.h>
#include <stdint.h>

// ---------------- problem constants (from reference) ----------------
#define SIZE_W 96
#define NPIX   (SIZE_W * SIZE_W)     // 9216 pixels per batch
#define NPTS   8192
#define NCHAN  64
#define NBATCH 2
#define KSEL   8
#define CHUNK  1024                  // points staged per LDS chunk
#define NCHUNK (NPTS / CHUNK)        // 8
#define R_NDC  (1.5f / 96.0f * 2.0f) // 0.03125
#define R2     (R_NDC * R_NDC)       // 9.765625e-4
#define QCAP   64                    // candidate queue slots per pixel
#define STAGE_FLOATS (2 * CHUNK * 3)
#define STAGE_BYTES  (STAGE_FLOATS * 4)          // 24576
#define SMEM_BYTES   (STAGE_BYTES + 256 * 4 + 256 * QCAP * 8) // 156672

typedef __attribute__((ext_vector_type(4))) unsigned int u32x4_t;
typedef __attribute__((ext_vector_type(8))) int          i32x8_t;
typedef __attribute__((ext_vector_type(4))) int          i32x4_t;
typedef __attribute__((ext_vector_type(2))) float        v2f;
typedef __attribute__((ext_vector_type(8))) float        v8f;

#if defined(__HIP_DEVICE_COMPILE__) && \
    __has_builtin(__builtin_amdgcn_tensor_load_to_lds) && \
    __has_builtin(__builtin_amdgcn_s_wait_tensorcnt)
#define USE_TDM 1
#else
#define USE_TDM 0
#endif

#if defined(__HIP_DEVICE_COMPILE__) && \
    __has_builtin(__builtin_amdgcn_wmma_f32_16x16x4_f32)
#define USE_WMMA 1
#else
#define USE_WMMA 0
#endif

#if USE_TDM
// 1-D DMA of `nfloats` f32 from global memory into LDS at byte offset
// `lds_off`. D# per CDNA5 ISA §8.
__device__ __forceinline__ void tdm_load_f32(uint32_t lds_off,
                                             const float* gptr,
                                             uint32_t nfloats) {
  uint64_t ga = (uint64_t)(uintptr_t)gptr;
  u32x4_t g0;
  g0[0] = 1u;                                   // count=1, user mode
  g0[1] = lds_off;                              // lds_addr (bytes)
  g0[2] = (uint32_t)ga;                         // global_addr[31:0]
  g0[3] = ((uint32_t)(ga >> 32) & 0x1FFFFFFu)   // global_addr[56:32]
          | (2u << 30);                         // type=2 ("image")
  i32x8_t g1;
  g1[0] = (int)(2u << 16);                      // data_size = 4 bytes
  g1[1] = (int)((nfloats & 0xFFFFu) << 16);     // tensor_dim0[15:0]
  g1[2] = (int)(nfloats >> 16);                 // tensor_dim0[31:16]
  g1[3] = (int)((nfloats & 0xFFFFu) << 16);     // tile_dim0 (1-D tile)
  g1[4] = 0;
  g1[5] = (int)nfloats;                         // tensor_dim0_stride lo
  g1[6] = 0;
  g1[7] = 0;
  i32x4_t z4 = {0, 0, 0, 0};
  i32x8_t z8 = {0, 0, 0, 0, 0, 0, 0, 0};
  __builtin_amdgcn_tensor_load_to_lds(g0, g1, z4, z4, z8, 0);
}
#endif

// ---------------- feature transpose: [B][C][N] -> [B][N][C] ----------------
__global__ __launch_bounds__(256) void transpose_feat(
    const float* __restrict__ src, float* __restrict__ ftr) {
  __shared__ float tile[16][17];
  const int b  = blockIdx.z;
  const int n0 = blockIdx.x * 16;
  const int c0 = blockIdx.y * 16;
  const int tx = threadIdx.x, ty = threadIdx.y;
  tile[ty][tx] = src[((size_t)b * NCHAN + c0 + ty) * NPTS + n0 + tx];
  __syncthreads();
  ftr[((size_t)b * NPTS + n0 + ty) * NCHAN + c0 + tx] = tile[tx][ty];
}

// ---------------- fused rasterize + top-8 + alpha blend ----------------
__global__ __launch_bounds__(256) void raster_blend(
    const float* __restrict__ pts,   // [B][N][3] raw (negation folded)
    const float* __restrict__ src,   // [B][C][N]
    const float* __restrict__ ftr,   // [B][N][C] (if use_tr)
    float* __restrict__ out,         // [B][C][P]
    int use_tr) {
  extern __shared__ char smraw[];
  float*        sm_pts = (float*)smraw;                       // staging
  unsigned int* cnt    = (unsigned int*)(smraw + STAGE_BYTES);
  float*        qz     = (float*)(smraw + STAGE_BYTES + 256 * 4);
  int*          qi     = (int*)(smraw + STAGE_BYTES + 256 * 4 + 256 * QCAP * 4);

  const int b  = blockIdx.x / (NPIX / 256);
  const int p0 = (blockIdx.x % (NPIX / 256)) * 256;   // first pixel of block
  const int p  = p0 + threadIdx.x;                    // this thread's pixel
  // pts = pts3D * [-1,-1,1]  =>  pix - pt = pix + raw_xy
  const float pix_x = 1.0f - (2.0f * (float)(p % SIZE_W) + 1.0f) / (float)SIZE_W;
  const float pix_y = 1.0f - (2.0f * (float)(p / SIZE_W) + 1.0f) / (float)SIZE_W;
  const float* gp = pts + (size_t)b * NPTS * 3;

  float zl[KSEL];
  int   il[KSEL];
#pragma unroll
  for (int k = 0; k < KSEL; ++k) { zl[k] = __builtin_huge_valf(); il[k] = 0; }

  cnt[threadIdx.x] = 0u;                              // queue counters

#if USE_WMMA
  // Per-wave WMMA operands: wave w owns row-tiles 2w, 2w+1 (pixels
  // p0+32w .. p0+32w+31).  A(16x4) row = [pix_x, pix_y, pix_sq - R2, 1].
  const int lane = threadIdx.x & 31;
  const int wid  = threadIdx.x >> 5;
  const int row  = lane & 15;
  const bool hi  = lane >= 16;                        // K=2,3 half
  v2f a0, a1;
  {
    const int q0 = p0 + (2 * wid) * 16 + row;
    const int q1 = q0 + 16;
    const float px0 = 1.0f - (2.0f * (float)(q0 % SIZE_W) + 1.0f) / (float)SIZE_W;
    const float py0 = 1.0f - (2.0f * (float)(q0 / SIZE_W) + 1.0f) / (float)SIZE_W;
    const float px1 = 1.0f - (2.0f * (float)(q1 % SIZE_W) + 1.0f) / (float)SIZE_W;
    const float py1 = 1.0f - (2.0f * (float)(q1 / SIZE_W) + 1.0f) / (float)SIZE_W;
    const float ps0 = __builtin_fmaf(px0, px0, py0 * py0) - R2;
    const float ps1 = __builtin_fmaf(px1, px1, py1 * py1) - R2;
    a0.x = hi ? ps0 : px0;  a0.y = hi ? 1.0f : py0;
    a1.x = hi ? ps1 : px1;  a1.y = hi ? 1.0f : py1;
  }
  const v8f czero = {0.f, 0.f, 0.f, 0.f, 0.f, 0.f, 0.f, 0.f};
#endif

#if USE_TDM
  if (threadIdx.x < 32) {            // wave 0 is the DMA producer
    tdm_load_f32(0u, gp, CHUNK * 3);
    __builtin_amdgcn_s_wait_tensorcnt(0);
  }
  __syncthreads();                   // covers cnt[] init as well
#else
  __syncthreads();
#endif

  for (int c = 0; c < NCHUNK; ++c) {
#if USE_TDM
    if (threadIdx.x < 32 && (c + 1) < NCHUNK)   // prefetch next chunk
      tdm_load_f32((uint32_t)(((c + 1) & 1) * (CHUNK * 3 * 4)),
                   gp + (size_t)(c + 1) * CHUNK * 3, CHUNK * 3);
    const float* buf = sm_pts + (c & 1) * (CHUNK * 3);
#else
    __syncthreads();
    for (int t = threadIdx.x; t < CHUNK * 3; t += 256)
      sm_pts[t] = gp[(size_t)c * CHUNK * 3 + t];
    __syncthreads();
    const float* buf = sm_pts;
#endif
    const int cbase = c * CHUNK;

#if USE_WMMA
    // D = A x B : D[p][n] = d2(p,n) - R2 ; hit iff D < 0.
    for (int g = 0; g < CHUNK / 16; ++g) {
      const float* pp = buf + 3 * (g * 16 + (lane & 15));
      const float X = pp[0], Y = pp[1], Z = pp[2];
      v2f bb;                                   // B col n = [2X, 2Y, 1, ptsq]
      bb.x = hi ? 1.0f : (2.0f * X);
      bb.y = hi ? ((Z > 0.0f) ? __builtin_fmaf(X, X, Y * Y) : 1.0e30f)
                : (2.0f * Y);
      v8f d0 = __builtin_amdgcn_wmma_f32_16x16x4_f32(
          false, a0, false, bb, (short)0, czero, false, false);
      v8f d1 = __builtin_amdgcn_wmma_f32_16x16x4_f32(
          false, a1, false, bb, (short)0, czero, false, false);
      const float m0 = fminf(fminf(fminf(d0[0], d0[1]), fminf(d0[2], d0[3])),
                             fminf(fminf(d0[4], d0[5]), fminf(d0[6], d0[7])));
      const float m1 = fminf(fminf(fminf(d1[0], d1[1]), fminf(d1[2], d1[3])),
                             fminf(fminf(d1[4], d1[5]), fminf(d1[6], d1[7])));
      if (fminf(m0, m1) < 0.0f) {               // rare: ~0.05 hits per wmma
        const int idxpt = cbase + g * 16 + (lane & 15);
        const int rbase = 32 * wid + (hi ? 8 : 0);
#pragma unroll
        for (int r = 0; r < 8; ++r) {
          if (d0[r] < 0.0f) {
            const int pl = rbase + r;           // row-tile 2w
            const unsigned s = atomicAdd(&cnt[pl], 1u);
            if (s < QCAP) { qz[pl * QCAP + s] = Z; qi[pl * QCAP + s] = idxpt; }
          }
          if (d1[r] < 0.0f) {
            const int pl = rbase + 16 + r;      // row-tile 2w+1
            const unsigned s = atomicAdd(&cnt[pl], 1u);
            if (s < QCAP) { qz[pl * QCAP + s] = Z; qi[pl * QCAP + s] = idxpt; }
          }
        }
      }
    }
#else
    // scalar fallback: per-thread scan with in-register top-8
#pragma unroll 4
    for (int j = 0; j < CHUNK; ++j) {
      const float X = buf[3 * j + 0];
      const float Y = buf[3 * j + 1];
      const float Z = buf[3 * j + 2];
      const float dx = pix_x + X;
      const float dy = pix_y + Y;
      const float d2 = __builtin_fmaf(dx, dx, dy * dy);
      if (d2 < R2 && Z > 0.0f && Z < zl[KSEL - 1]) {
        float zz = Z; int ii = cbase + j;
#pragma unroll
        for (int k = 0; k < KSEL; ++k) {
          const bool  sw = zz < zl[k];
          const float tz = zl[k]; const int ti = il[k];
          zl[k] = sw ? zz : zl[k];  il[k] = sw ? ii : il[k];
          zz    = sw ? tz : zz;     ii    = sw ? ti : ii;
        }
      }
    }
#endif

#if USE_TDM
    if (threadIdx.x < 32 && (c + 1) < NCHUNK)
      __builtin_amdgcn_s_wait_tensorcnt(0);     // next buffer landed
    __syncthreads();                            // release buffers/queues
#endif
  }
  __syncthreads();                              // all queue pushes visible

#if USE_WMMA
  // drain this pixel's candidate queue into the top-8 list;
  // lexicographic (z, idx) compare reproduces top_k tie ordering.
  {
    int cn = (int)cnt[threadIdx.x];
    cn = cn > QCAP ? QCAP : cn;
    for (int e = 0; e < cn; ++e) {
      float zz = qz[threadIdx.x * QCAP + e];
      int   ii = qi[threadIdx.x * QCAP + e];
      if (zz < zl[KSEL - 1] || (zz == zl[KSEL - 1] && ii < il[KSEL - 1])) {
#pragma unroll
        for (int k = 0; k < KSEL; ++k) {
          const bool sw = (zz < zl[k]) || (zz == zl[k] && ii < il[k]);
          const float tz = zl[k]; const int ti = il[k];
          zl[k] = sw ? zz : zl[k];  il[k] = sw ? ii : il[k];
          zz    = sw ? tz : zz;     ii    = sw ? ti : ii;
        }
      }
    }
  }
#endif

  // ---- front-to-back compositing weights ----
  float wgt[KSEL];
  float T = 1.0f;
#pragma unroll
  for (int k = 0; k < KSEL; ++k) {
    float a = 0.0f;
    if (zl[k] < 3.0e38f) {
      const float X = gp[(size_t)il[k] * 3 + 0];
      const float Y = gp[(size_t)il[k] * 3 + 1];
      const float dx = pix_x + X;
      const float dy = pix_y + Y;
      float dist = __builtin_fmaf(dx, dx, dy * dy) * (1.0f / R2);
      dist = fminf(fmaxf(dist, 0.001f), 1.0f);
      a = 1.0f - __builtin_sqrtf(dist);          // TAU == 1
    }
    wgt[k] = a * T;
    T *= (1.0f - a);
  }

  // ---- channel accumulation: 4 chunks of 16 channels ----
  float* op = out + (size_t)b * NCHAN * NPIX + p;
  for (int cc = 0; cc < NCHAN; cc += 16) {
    float acc[16];
#pragma unroll
    for (int i = 0; i < 16; ++i) acc[i] = 0.0f;
#pragma unroll
    for (int k = 0; k < KSEL; ++k) {
      const float w = wgt[k];
      if (w > 0.0f) {
        if (use_tr) {                            // 256B-contiguous row gather
          const float4* r = (const float4*)(ftr +
              ((size_t)b * NPTS + il[k]) * NCHAN + cc);
#pragma unroll
          for (int q = 0; q < 4; ++q) {
            const float4 f = r[q];
            acc[4 * q + 0] = __builtin_fmaf(w, f.x, acc[4 * q + 0]);
            acc[4 * q + 1] = __builtin_fmaf(w, f.y, acc[4 * q + 1]);
            acc[4 * q + 2] = __builtin_fmaf(w, f.z, acc[4 * q + 2]);
            acc[4 * q + 3] = __builtin_fmaf(w, f.w, acc[4 * q + 3]);
          }
        } else {                                 // strided fallback
#pragma unroll
          for (int i = 0; i < 16; ++i)
            acc[i] = __builtin_fmaf(
                w, src[((size_t)b * NCHAN + cc + i) * NPTS + il[k]], acc[i]);
        }
      }
    }
#pragma unroll
    for (int i = 0; i < 16; ++i)
      op[(size_t)(cc + i) * NPIX] = acc[i];      // coalesced per channel plane
  }
}

// ---------------- host launcher ----------------
extern "C" void kernel_launch(void* const* d_in, const int* in_sizes, int n_in,
                              void* d_out, int out_size, void* d_ws,
                              size_t ws_size, hipStream_t stream) {
  const float* pts = (const float*)d_in[0];  // (2, 8192, 3) f32
  const float* src = (const float*)d_in[1];  // (2, 64, 8192) f32
  float* out = (float*)d_out;                // (2, 64, 96, 96) f32

  const size_t need = (size_t)NBATCH * NPTS * NCHAN * sizeof(float); // 4 MB
  const int use_tr = (d_ws != nullptr && ws_size >= need) ? 1 : 0;
  float* ftr = (float*)d_ws;

  if (use_tr) {
    dim3 g(NPTS / 16, NCHAN / 16, NBATCH), blk(16, 16);
    hipLaunchKernelGGL(transpose_feat, g, blk, 0, stream, src, ftr);
  }
  // allow >64KB dynamic LDS (WGP has 320KB on CDNA5)
  (void)hipFuncSetAttribute((const void*)raster_blend,
                            hipFuncAttributeMaxDynamicSharedMemorySize,
                            SMEM_BYTES);
  dim3 g2((NPIX / 256) * NBATCH);              // 72 blocks, 8 waves each
  hipLaunchKernelGGL(raster_blend, g2, dim3(256), SMEM_BYTES, stream,
                     pts, src, ftr, out, use_tr);
}